// MotionBasedCompression_77773267796179
// MI455X (gfx1250) — compile-verified
//
#include <hip/hip_runtime.h>
#include <stdint.h>

// Problem constants (from setup_inputs: motion (512,1,224,224) f32, num_tokens=6272)
#define NB     512
#define IMG    224
#define HP     14
#define TPP    32
#define NCELL  (NB * HP * HP)          // 512*196 = 100352
#define TILE_R 16                       // rows per TDM tile == pooling window height
#define TILE_F (TILE_R * IMG)           // floats per tile = 3584 (14336 B)

typedef __attribute__((ext_vector_type(4))) unsigned int u32x4_t;
typedef __attribute__((ext_vector_type(8))) int          i32x8_t;
typedef __attribute__((ext_vector_type(4))) int          i32x4_t;

// ---------------------------------------------------------------------------
// Tensor Data Mover: async DMA of a 2-D f32 tile (width x rows) from global
// memory into LDS.  D# layout per CDNA5 ISA.
// ---------------------------------------------------------------------------
__device__ __forceinline__ void tdm_load_tile_f32(uint32_t lds_byte_off,
                                                  uint64_t gaddr,
                                                  uint32_t width_elems,
                                                  uint32_t rows,
                                                  uint32_t row_stride_elems) {
  u32x4_t g0;
  g0.x = 1u;                                           // count=1, user D#, no gather
  g0.y = lds_byte_off;                                 // lds_addr [63:32]
  g0.z = (uint32_t)(gaddr & 0xFFFFFFFFu);              // global_addr[31:0]
  g0.w = (uint32_t)((gaddr >> 32) & 0x01FFFFFFu)       // global_addr[56:32]
         | (2u << 30);                                 // type=2 ("image")

  i32x8_t g1;
  g1[0] = (int)(2u << 16);                             // data_size=2 -> 4 bytes
  g1[1] = (int)(width_elems << 16);                    // tensor_dim0[15:0] @ [63:48]
  g1[2] = (int)((width_elems >> 16) | (rows << 16));   // dim0 hi / tensor_dim1 lo
  g1[3] = (int)((rows >> 16) | (width_elems << 16));   // dim1 hi / tile_dim0
  g1[4] = (int)rows;                                   // tile_dim1 (tile_dim2=0 -> 2D)
  g1[5] = (int)row_stride_elems;                       // tensor_dim0_stride[31:0]
  g1[6] = 0;                                           // stride hi / dim1_stride lo
  g1[7] = 0;

  i32x4_t gz = (i32x4_t)0;                             // groups 2/3 unused (2-D)
#if defined(__clang_major__) && (__clang_major__ >= 23)
  i32x8_t gz8 = (i32x8_t)0;
  __builtin_amdgcn_tensor_load_to_lds(g0, g1, gz, gz, gz8, 0);
#else
  __builtin_amdgcn_tensor_load_to_lds(g0, g1, gz, gz, 0);
#endif
}

// ---------------------------------------------------------------------------
// K1: one workgroup per image.  TDM double-buffers 16-row tiles into LDS;
// each tile is exactly one pooled output row (fh=16).  Produces pooled[512*196]
// plus per-image min/max/sum partials.
// ---------------------------------------------------------------------------
__global__ __launch_bounds__(256) void pool_tdm_kernel(
    const float* __restrict__ motion,
    float* __restrict__ pooled,
    float* __restrict__ bmin, float* __restrict__ bmax, float* __restrict__ bsum) {
  __shared__ float tile[2][TILE_F];     // 2 x 14336 B double buffer
  __shared__ float partial[TILE_R * HP];
  __shared__ float redmn[HP], redmx[HP], redsm[HP];

  const int img = blockIdx.x;
  const int th  = threadIdx.x;
  const bool wave0 = (th < 32);         // wave-uniform on wave32

  const uint64_t gbase = (uint64_t)(uintptr_t)(motion + (size_t)img * IMG * IMG);
  // Generic pointer to __shared__: low 32 bits carry the LDS byte offset
  // (LDS aperture signature lives in [63:32]).
  const uint32_t ldsA = (uint32_t)(uintptr_t)(&tile[0][0]);
  const uint32_t ldsB = (uint32_t)(uintptr_t)(&tile[1][0]);

  if (wave0) {                          // prologue: DMA tile 0 -> buf A
    tdm_load_tile_f32(ldsA, gbase, IMG, TILE_R, IMG);
  }

  float lmn = 3.402823466e38f, lmx = -3.402823466e38f, lsm = 0.0f;

  for (int t = 0; t < HP; ++t) {
    const float* cur = (t & 1) ? &tile[1][0] : &tile[0][0];
    if (wave0) {
      if (t + 1 < HP) {                 // prefetch next tile into other buffer
        tdm_load_tile_f32(((t + 1) & 1) ? ldsB : ldsA,
                          gbase + (uint64_t)(t + 1) * TILE_R * IMG * 4ull,
                          IMG, TILE_R, IMG);
        __builtin_amdgcn_s_wait_tensorcnt(1);  // tile t done (in-order), t+1 in flight
      } else {
        __builtin_amdgcn_s_wait_tensorcnt(0);
      }
    }
    __syncthreads();                    // tile t visible in LDS to all waves

    // stage 1: 224 row-segment sums (16 contiguous floats each, 4x ds b128)
    if (th < TILE_R * HP) {
      const int r = th / HP, c = th % HP;
      const float4* p = reinterpret_cast<const float4*>(cur + r * IMG + c * 16);
      float4 a = p[0], b = p[1], c4 = p[2], d = p[3];
      partial[th] = (a.x + a.y + a.z + a.w) + (b.x + b.y + b.z + b.w) +
                    (c4.x + c4.y + c4.z + c4.w) + (d.x + d.y + d.z + d.w);
    }
    __syncthreads();

    // stage 2: fold 16 rows -> pooled row t
    if (th < HP) {
      float s = 0.0f;
      #pragma unroll
      for (int r = 0; r < TILE_R; ++r) s += partial[r * HP + th];
      s *= (1.0f / 256.0f);
      pooled[(size_t)img * (HP * HP) + t * HP + th] = s;
      lmn = fminf(lmn, s); lmx = fmaxf(lmx, s); lsm += s;
    }
    __syncthreads();                    // partial[] + consumed buffer free for reuse
  }

  if (th < HP) { redmn[th] = lmn; redmx[th] = lmx; redsm[th] = lsm; }
  __syncthreads();
  if (th == 0) {
    float mn = redmn[0], mx = redmx[0], sm = redsm[0];
    #pragma unroll
    for (int i = 1; i < HP; ++i) {
      mn = fminf(mn, redmn[i]); mx = fmaxf(mx, redmx[i]); sm += redsm[i];
    }
    bmin[img] = mn; bmax[img] = mx; bsum[img] = sm;
  }
}

// ---------------------------------------------------------------------------
// K2: single block reduces the 512 per-image partials and derives the four
// broadcast scalars: m_min, 1/(range+1e-6), scale, avg.
// ---------------------------------------------------------------------------
__global__ __launch_bounds__(256) void reduce_kernel(
    const float* __restrict__ bmin, const float* __restrict__ bmax,
    const float* __restrict__ bsum, float* __restrict__ scalars) {
  __shared__ float smn[256], smx[256], ssm[256];
  const int th = threadIdx.x;
  smn[th] = fminf(bmin[th], bmin[th + 256]);
  smx[th] = fmaxf(bmax[th], bmax[th + 256]);
  ssm[th] = bsum[th] + bsum[th + 256];
  __syncthreads();
  for (int s = 128; s > 0; s >>= 1) {
    if (th < s) {
      smn[th] = fminf(smn[th], smn[th + s]);
      smx[th] = fmaxf(smx[th], smx[th + s]);
      ssm[th] += ssm[th + s];
    }
    __syncthreads();
  }
  if (th == 0) {
    const float m_min = smn[0], m_max = smx[0], total = ssm[0];
    const float inv   = 1.0f / (m_max - m_min + 1e-6f);
    const float mean_norm = (total / (float)NCELL - m_min) * inv;
    const float avg   = 0.25f + 0.75f * mean_norm;   // mean of affine == affine of mean
    const float scale = 0.5f / (avg + 1e-6f);
    scalars[0] = m_min; scalars[1] = inv; scalars[2] = scale; scalars[3] = avg;
  }
}

// ---------------------------------------------------------------------------
// K3: one wave per cell, lane j = token index.  Lane stores are a perfectly
// coalesced 128 B line per wave.  rintf == round-half-even == jnp.round.
// ---------------------------------------------------------------------------
__global__ __launch_bounds__(256) void mask_kernel(
    const float* __restrict__ pooled,
    const float* __restrict__ scalars,
    float* __restrict__ out) {
  __shared__ float s4[4];
  if (threadIdx.x < 4) s4[threadIdx.x] = scalars[threadIdx.x];
  __syncthreads();

  const int gid  = blockIdx.x * 256 + threadIdx.x;
  const int cell = gid >> 5;
  const int j    = gid & 31;
  if (cell >= NCELL) return;

  const float m_min = s4[0], inv = s4[1], scale = s4[2], avg = s4[3];
  const float p      = pooled[cell];           // same addr per wave -> one request
  const float norm   = (p - m_min) * inv;
  const float ratio  = 0.25f + 0.75f * norm;
  const float scaled = fminf(fmaxf(ratio * scale, 0.25f), 1.0f);
  const float finalr = (avg > 0.0f) ? scaled : ratio;

  int k = (int)rintf(finalr * (float)TPP);
  k = (k < 1) ? 1 : (k > TPP ? TPP : k);

  out[(size_t)2 * NCELL + (size_t)cell * TPP + j] = (j < k) ? 1.0f : 0.0f;
  if (j == 0) {
    out[cell]         = finalr;                // compression_ratio
    out[NCELL + cell] = norm;                  // importance_scores
  }
}

// ---------------------------------------------------------------------------
extern "C" void kernel_launch(void* const* d_in, const int* in_sizes, int n_in,
                              void* d_out, int out_size, void* d_ws, size_t ws_size,
                              hipStream_t stream) {
  const float* motion = (const float*)d_in[0];
  float* out = (float*)d_out;

  // workspace layout (floats): pooled[100352] | bmin[512] | bmax[512] | bsum[512] | scalars[4]
  float* ws      = (float*)d_ws;
  float* pooled  = ws;
  float* bmin    = pooled + NCELL;
  float* bmax    = bmin + NB;
  float* bsum    = bmax + NB;
  float* scalars = bsum + NB;

  pool_tdm_kernel<<<NB, 256, 0, stream>>>(motion, pooled, bmin, bmax, bsum);
  reduce_kernel<<<1, 256, 0, stream>>>(bmin, bmax, bsum, scalars);
  mask_kernel<<<(NCELL * TPP) / 256, 256, 0, stream>>>(pooled, scalars, out);
}